// FConv2d_79877801771392
// MI455X (gfx1250) — compile-verified
//
#include <hip/hip_runtime.h>

// ---------------------------------------------------------------------------
// FNO spectral conv for MI455X (gfx1250): every DFT stage expressed as dense
// matmul on V_WMMA_F32_16X16X4_F32 tiles (wave32). Whole working set (~87MB)
// is L2-resident (192MB), so f32 WMMA is the right precision: problem is
// neither flop- nor bandwidth-bound at this size.
// ---------------------------------------------------------------------------

typedef __attribute__((ext_vector_type(2))) float v2f;
typedef __attribute__((ext_vector_type(8))) float v8f;

constexpr int NB  = 16;          // batch
constexpr int NC  = 64;          // cin
constexpr int NFR = 8;           // n (modes)
constexpr int ND  = 16;          // d (reduced channels)
constexpr int NL  = 64;          // spatial
constexpr int NLP = 65;          // padded spatial
constexpr int NP  = NLP * NLP;   // 4225 freq points
constexpr int NPW = NLP * NL;    // 4160 (P intermediate plane)

// Twiddle-table offsets (floats) inside workspace. Tables zero-padded to
// tile-friendly dims so WMMA K/M/N overruns multiply by exact zeros.
constexpr int OFF_FWDR = 0;                       // e^{-2pi i ab/65}, [80][80]
constexpr int OFF_FWDI = OFF_FWDR + 80 * 80;
constexpr int OFF_ER   = OFF_FWDI + 80 * 80;      // e^{-2pi i ck/64}, [64][64]
constexpr int OFF_EI   = OFF_ER + 64 * 64;
constexpr int OFF_WNR  = OFF_EI + 64 * 64;        // e^{+2pi i c*4d/64}/64, [64][16]
constexpr int OFF_WNI  = OFF_WNR + 64 * 16;
constexpr int OFF_IVR  = OFF_WNI + 64 * 16;       // e^{+2pi i v(w+1)/65}, [68][64]
constexpr int OFF_IVI  = OFF_IVR + 68 * 64;
constexpr int OFF_IUR  = OFF_IVI + 68 * 64;       // e^{+2pi i u(h+1)/65}/4225, [64][68]
constexpr int OFF_IUI  = OFF_IUR + 64 * 68;
constexpr int TW_TOTAL = OFF_IUI + 64 * 68;       // 40448 floats
constexpr int TW_PAD   = 40960;

#define PI_D 3.14159265358979323846

// ---- WMMA helpers (16x16x4 f32, wave32) -----------------------------------

static __device__ __forceinline__ v8f wmma4(v2f a, v2f b, v8f c) {
  // (neg_a, A, neg_b, B, c_mod, C, reuse_a, reuse_b); f32 NEG only supports
  // C-negate per ISA, so pass false and negate A in VALU when needed.
  return __builtin_amdgcn_wmma_f32_16x16x4_f32(false, a, false, b, (short)0, c,
                                               false, false);
}

static __device__ __forceinline__ v8f zero8() {
  v8f z = {0.f, 0.f, 0.f, 0.f, 0.f, 0.f, 0.f, 0.f};
  return z;
}

// A fragment: A[m][k] row-major, lanes 0-15 -> M rows, K split {0,1|2,3}.
static __device__ __forceinline__ v2f frag_a(const float* __restrict__ A,
                                             int lda, int m0, int k0) {
  int lane = threadIdx.x & 31;
  int m = m0 + (lane & 15);
  int k = k0 + ((lane >> 4) << 1);
  const float* p = A + m * lda + k;
  v2f r;
  r.x = p[0];
  r.y = p[1];
  return r;
}

static __device__ __forceinline__ v2f frag_a_guard(const float* __restrict__ A,
                                                   int lda, int m0, int k0,
                                                   int mmax, int kmax) {
  int lane = threadIdx.x & 31;
  int m = m0 + (lane & 15);
  int k = k0 + ((lane >> 4) << 1);
  v2f r;
  r.x = 0.f;
  r.y = 0.f;
  if (m < mmax) {
    const float* p = A + m * lda + k;
    if (k < kmax) r.x = p[0];
    if (k + 1 < kmax) r.y = p[1];
  }
  return r;
}

// B fragment: B[k][n] row-major, lanes 0-15 -> N cols with K{0,1}, 16-31 K{2,3}.
static __device__ __forceinline__ v2f frag_b(const float* __restrict__ B,
                                             int ldb, int k0, int n0) {
  int lane = threadIdx.x & 31;
  int n = n0 + (lane & 15);
  int k = k0 + ((lane >> 4) << 1);
  const float* p = B + k * ldb + n;
  v2f r;
  r.x = p[0];
  r.y = p[ldb];
  return r;
}

static __device__ __forceinline__ v2f frag_b_nguard(const float* __restrict__ B,
                                                    int ldb, int k0, int n0,
                                                    int nmax) {
  int lane = threadIdx.x & 31;
  int n = n0 + (lane & 15);
  int k = k0 + ((lane >> 4) << 1);
  v2f r;
  r.x = 0.f;
  r.y = 0.f;
  if (n < nmax) {
    const float* p = B + k * ldb + n;
    r.x = p[0];
    r.y = p[ldb];
  }
  return r;
}

static __device__ __forceinline__ v2f frag_b_kguard(const float* __restrict__ B,
                                                    int ldb, int k0, int n0,
                                                    int kmax) {
  int lane = threadIdx.x & 31;
  int n = n0 + (lane & 15);
  int k = k0 + ((lane >> 4) << 1);
  v2f r;
  r.x = 0.f;
  r.y = 0.f;
  const float* p = B + k * ldb + n;
  if (k < kmax) r.x = p[0];
  if (k + 1 < kmax) r.y = p[ldb];
  return r;
}

// Complex MAC: dr += ar*br - ai*bi ; di += ar*bi + ai*br   (4 WMMAs)
static __device__ __forceinline__ void cwmma(v2f ar, v2f ai, v2f br, v2f bi,
                                             v8f& dr, v8f& di) {
  dr = wmma4(ar, br, dr);
  v2f nai;
  nai.x = -ai.x;
  nai.y = -ai.y;
  dr = wmma4(nai, bi, dr);
  di = wmma4(ar, bi, di);
  di = wmma4(ai, br, di);
}

// ---- Kernel 0: twiddle tables ---------------------------------------------

__global__ void k_twiddles(float* __restrict__ tw) {
  int t = blockIdx.x * blockDim.x + threadIdx.x;
  int which = blockIdx.y;
  if (which == 0) {                       // forward DFT-65, [80][80] zero-pad
    if (t >= 80 * 80) return;
    int a = t / 80, b = t % 80;
    float r = 0.f, im = 0.f;
    if (a < NLP && b < NLP) {
      double ang = -2.0 * PI_D * (double)((a * b) % NLP) / (double)NLP;
      r = (float)cos(ang);
      im = (float)sin(ang);
    }
    tw[OFF_FWDR + t] = r;
    tw[OFF_FWDI + t] = im;
  } else if (which == 1) {                // channel DFT-64
    if (t >= NC * NC) return;
    int c = t / NC, k = t % NC;
    double ang = -2.0 * PI_D * (double)((c * k) % NC) / (double)NC;
    tw[OFF_ER + t] = (float)cos(ang);
    tw[OFF_EI + t] = (float)sin(ang);
  } else if (which == 2) {                // WN[c][d] = e^{+2pi i c*4d/64}/64
    if (t >= NC * ND) return;
    int c = t / ND, d = t % ND;
    double ang = 2.0 * PI_D * (double)((c * 4 * d) % NC) / (double)NC;
    tw[OFF_WNR + t] = (float)(cos(ang) / 64.0);
    tw[OFF_WNI + t] = (float)(sin(ang) / 64.0);
  } else if (which == 3) {                // IV[v][w] = e^{+2pi i v(w+1)/65}
    if (t >= 68 * NL) return;
    int v = t / NL, w = t % NL;
    float r = 0.f, im = 0.f;
    if (v < NLP) {
      double ang = 2.0 * PI_D * (double)((v * (w + 1)) % NLP) / (double)NLP;
      r = (float)cos(ang);
      im = (float)sin(ang);
    }
    tw[OFF_IVR + t] = r;
    tw[OFF_IVI + t] = im;
  } else {                                // IU[h][u] = e^{+2pi i u(h+1)/65}/65^2
    if (t >= NL * 68) return;
    int h = t / 68, u = t % 68;
    float r = 0.f, im = 0.f;
    if (u < NLP) {
      double ang = 2.0 * PI_D * (double)((u * (h + 1)) % NLP) / (double)NLP;
      double s = 1.0 / (double)(NLP * NLP);
      r = (float)(cos(ang) * s);
      im = (float)(sin(ang) * s);
    }
    tw[OFF_IUR + t] = r;
    tw[OFF_IUI + t] = im;
  }
}

// ---- Kernel 1: fused 2D spatial DFT per (b,c) plane (x real -> Y complex) --

__global__ __launch_bounds__(128) void k_fft2d(const float* __restrict__ x,
                                               const float* __restrict__ tw,
                                               float* __restrict__ Yr,
                                               float* __restrict__ Yi) {
  __shared__ float xs[NL * NL];       // 16 KB
  __shared__ float Tr[NL * 80];       // 20 KB
  __shared__ float Ti[NL * 80];       // 20 KB
  const int plane = blockIdx.x;       // b*64 + c
  const int tid = threadIdx.x;
  const int wave = tid >> 5;
  const int lane = tid & 31;

  const float* xp = x + (size_t)plane * (NL * NL);
  for (int i = tid; i < NL * NL; i += 128) xs[i] = xp[i];
  __syncthreads();

  const float* Fr = tw + OFF_FWDR;
  const float* Fi = tw + OFF_FWDI;

  // Stage 1: T(h,v) = sum_w xs[h][w] * F[w][v]   (x real)
  for (int t = wave; t < 4 * 5; t += 4) {
    int mo = (t / 5) * 16, no = (t % 5) * 16;
    v8f ar = zero8(), ai = zero8();
    for (int k0 = 0; k0 < NL; k0 += 4) {
      v2f a = frag_a(xs, NL, mo, k0);
      ar = wmma4(a, frag_b(Fr, 80, k0, no), ar);
      ai = wmma4(a, frag_b(Fi, 80, k0, no), ai);
    }
    int n = no + (lane & 15);
    int mb = mo + ((lane >> 4) << 3);
#pragma unroll
    for (int i = 0; i < 8; ++i) {
      Tr[(mb + i) * 80 + n] = ar[i];
      Ti[(mb + i) * 80 + n] = ai[i];
    }
  }
  __syncthreads();

  // Stage 2: Y(u,v) = sum_h F[u][h] * T(h,v)
  float* yr = Yr + (size_t)plane * NP;
  float* yi = Yi + (size_t)plane * NP;
  for (int t = wave; t < 5 * 5; t += 4) {
    int mo = (t / 5) * 16, no = (t % 5) * 16;
    v8f dr = zero8(), di = zero8();
    for (int k0 = 0; k0 < NL; k0 += 4) {
      v2f par = frag_a(Fr, 80, mo, k0);
      v2f pai = frag_a(Fi, 80, mo, k0);
      v2f pbr = frag_b(Tr, 80, k0, no);
      v2f pbi = frag_b(Ti, 80, k0, no);
      cwmma(par, pai, pbr, pbi, dr, di);
    }
    int n = no + (lane & 15);
    int mb = mo + ((lane >> 4) << 3);
    if (n < NLP) {
#pragma unroll
      for (int i = 0; i < 8; ++i) {
        int m = mb + i;
        if (m < NLP) {
          yr[m * NLP + n] = dr[i];
          yi[m * NLP + n] = di[i];
        }
      }
    }
  }
}

// ---- Kernel 2: channel DFT  X(b,c,p) = sum_k E[c][k] * Y(b,k,p) -----------

__global__ __launch_bounds__(128) void k_cfft(const float* __restrict__ tw,
                                              const float* __restrict__ Yr,
                                              const float* __restrict__ Yi,
                                              float* __restrict__ Xr,
                                              float* __restrict__ Xi) {
  const int b = blockIdx.y;
  const int no = blockIdx.x * 16;          // p tile
  const int mo = (threadIdx.x >> 5) * 16;  // c tile (4 waves = 4 tiles)
  const int lane = threadIdx.x & 31;
  const float* Er = tw + OFF_ER;
  const float* Ei = tw + OFF_EI;
  const float* yr = Yr + (size_t)b * NC * NP;
  const float* yi = Yi + (size_t)b * NC * NP;
  v8f dr = zero8(), di = zero8();
  for (int k0 = 0; k0 < NC; k0 += 4) {
    v2f ar = frag_a(Er, NC, mo, k0);
    v2f ai = frag_a(Ei, NC, mo, k0);
    v2f br = frag_b_nguard(yr, NP, k0, no, NP);
    v2f bi = frag_b_nguard(yi, NP, k0, no, NP);
    cwmma(ar, ai, br, bi, dr, di);
  }
  int n = no + (lane & 15);
  int mb = mo + ((lane >> 4) << 3);
  if (n < NP) {
    float* xr = Xr + (size_t)b * NC * NP;
    float* xi = Xi + (size_t)b * NC * NP;
#pragma unroll
    for (int i = 0; i < 8; ++i) {
      xr[(mb + i) * NP + n] = dr[i];
      xi[(mb + i) * NP + n] = di[i];
    }
  }
}

// ---- Kernel 3: per-point mode mix  G(b,d,p) = sum_c X*Wn * WN[c][d] -------

__global__ __launch_bounds__(128) void k_mix(const float* __restrict__ tw,
                                             const float* __restrict__ Xr,
                                             const float* __restrict__ Xi,
                                             const float* __restrict__ wre,
                                             const float* __restrict__ wim,
                                             float* __restrict__ Gr,
                                             float* __restrict__ Gi, int nidx) {
  const int wave = threadIdx.x >> 5;
  const int p = blockIdx.x * 4 + wave;   // one freq point per wave
  if (p >= NP) return;
  const int lane = threadIdx.x & 31;
  const float* WNr = tw + OFF_WNR;
  const float* WNi = tw + OFF_WNI;
  const int bm = lane & 15;              // A row = batch index
  const int khalf = (lane >> 4) << 1;
  v8f dr = zero8(), di = zero8();
  for (int k0 = 0; k0 < NC; k0 += 4) {
    v2f ar, ai;
#pragma unroll
    for (int j = 0; j < 2; ++j) {
      int c = k0 + khalf + j;
      size_t xo = ((size_t)bm * NC + c) * NP + p;
      size_t wo = ((size_t)nidx * NC + c) * NP + p;
      float xr = Xr[xo], xi = Xi[xo];
      float wr = wre[wo], wi = wim[wo];
      ar[j] = xr * wr - xi * wi;
      ai[j] = xr * wi + xi * wr;
    }
    v2f br = frag_b(WNr, ND, k0, 0);
    v2f bi = frag_b(WNi, ND, k0, 0);
    cwmma(ar, ai, br, bi, dr, di);
  }
  int n = lane & 15;                     // d
  int mb = (lane >> 4) << 3;             // batch rows
#pragma unroll
  for (int i = 0; i < 8; ++i) {
    size_t o = ((size_t)(mb + i) * ND + n) * NP + p;
    Gr[o] = dr[i];
    Gi[o] = di[i];
  }
}

// ---- Kernel 4: inverse DFT over v:  P(u,w) = sum_v G(u,v) * IV[v][w] ------

__global__ __launch_bounds__(128) void k_ifft_v(const float* __restrict__ tw,
                                                const float* __restrict__ Gr,
                                                const float* __restrict__ Gi,
                                                float* __restrict__ Pr,
                                                float* __restrict__ Pi) {
  const int d = blockIdx.x, b = blockIdx.y;
  const int wave = threadIdx.x >> 5;
  const int lane = threadIdx.x & 31;
  const float* IVr = tw + OFF_IVR;
  const float* IVi = tw + OFF_IVI;
  const float* gr = Gr + ((size_t)b * ND + d) * NP;
  const float* gi = Gi + ((size_t)b * ND + d) * NP;
  float* pr = Pr + ((size_t)b * ND + d) * NPW;
  float* pi = Pi + ((size_t)b * ND + d) * NPW;
  for (int t = wave; t < 5 * 4; t += 4) {
    int mo = (t / 4) * 16, no = (t % 4) * 16;
    v8f dr = zero8(), di = zero8();
    for (int k0 = 0; k0 < 68; k0 += 4) {
      v2f ar = frag_a_guard(gr, NLP, mo, k0, NLP, NLP);
      v2f ai = frag_a_guard(gi, NLP, mo, k0, NLP, NLP);
      v2f br = frag_b(IVr, NL, k0, no);
      v2f bi = frag_b(IVi, NL, k0, no);
      cwmma(ar, ai, br, bi, dr, di);
    }
    int n = no + (lane & 15);
    int mb = mo + ((lane >> 4) << 3);
#pragma unroll
    for (int i = 0; i < 8; ++i) {
      int m = mb + i;
      if (m < NLP) {
        pr[m * NL + n] = dr[i];
        pi[m * NL + n] = di[i];
      }
    }
  }
}

// ---- Kernel 5: inverse DFT over u, real part only, write output -----------

__global__ __launch_bounds__(128) void k_ifft_u(const float* __restrict__ tw,
                                                const float* __restrict__ Pr,
                                                const float* __restrict__ Pi,
                                                float* __restrict__ out,
                                                int nidx) {
  const int d = blockIdx.x, b = blockIdx.y;
  const int wave = threadIdx.x >> 5;
  const int lane = threadIdx.x & 31;
  const float* IUr = tw + OFF_IUR;
  const float* IUi = tw + OFF_IUI;
  const float* pr = Pr + ((size_t)b * ND + d) * NPW;
  const float* pi = Pi + ((size_t)b * ND + d) * NPW;
  float* op = out + ((size_t)b * (ND * NFR) + (d * NFR + nidx)) * (NL * NL);
  for (int t = wave; t < 4 * 4; t += 4) {
    int mo = (t / 4) * 16, no = (t % 4) * 16;
    v8f dr = zero8();
    for (int k0 = 0; k0 < 68; k0 += 4) {
      v2f ar = frag_a(IUr, 68, mo, k0);
      v2f ai = frag_a(IUi, 68, mo, k0);
      v2f br = frag_b_kguard(pr, NL, k0, no, NLP);
      v2f bi = frag_b_kguard(pi, NL, k0, no, NLP);
      dr = wmma4(ar, br, dr);            // + Re*Re
      v2f nai;
      nai.x = -ai.x;
      nai.y = -ai.y;
      dr = wmma4(nai, bi, dr);           // - Im*Im  -> real part
    }
    int n = no + (lane & 15);
    int mb = mo + ((lane >> 4) << 3);
#pragma unroll
    for (int i = 0; i < 8; ++i) op[(mb + i) * NL + n] = dr[i];
  }
}

// ---------------------------------------------------------------------------

extern "C" void kernel_launch(void* const* d_in, const int* in_sizes, int n_in,
                              void* d_out, int out_size, void* d_ws,
                              size_t ws_size, hipStream_t stream) {
  (void)in_sizes; (void)n_in; (void)out_size; (void)ws_size;
  const float* x   = (const float*)d_in[0];
  const float* wre = (const float*)d_in[1];
  const float* wim = (const float*)d_in[2];
  float* out = (float*)d_out;
  float* ws = (float*)d_ws;

  float* tw = ws;
  size_t off = TW_PAD;
  float* Yr = ws + off; off += (size_t)NB * NC * NP;
  float* Yi = ws + off; off += (size_t)NB * NC * NP;
  float* Xr = ws + off; off += (size_t)NB * NC * NP;
  float* Xi = ws + off; off += (size_t)NB * NC * NP;
  float* Gr = ws + off; off += (size_t)NB * ND * NP;   // reused per n
  float* Gi = ws + off; off += (size_t)NB * ND * NP;
  float* Pr = ws + off; off += (size_t)NB * ND * NPW;  // reused per n
  float* Pi = ws + off; off += (size_t)NB * ND * NPW;

  k_twiddles<<<dim3(25, 5), 256, 0, stream>>>(tw);
  k_fft2d<<<NB * NC, 128, 0, stream>>>(x, tw, Yr, Yi);
  k_cfft<<<dim3((NP + 15) / 16, NB), 128, 0, stream>>>(tw, Yr, Yi, Xr, Xi);
  for (int n = 0; n < NFR; ++n) {
    k_mix<<<(NP + 3) / 4, 128, 0, stream>>>(tw, Xr, Xi, wre, wim, Gr, Gi, n);
    k_ifft_v<<<dim3(ND, NB), 128, 0, stream>>>(tw, Gr, Gi, Pr, Pi);
    k_ifft_u<<<dim3(ND, NB), 128, 0, stream>>>(tw, Pr, Pi, out, n);
  }
}